// TDARetinanetFull_28681791602861
// MI455X (gfx1250) — compile-verified
//
#include <hip/hip_runtime.h>
#include <hip/hip_bf16.h>
#include <stdint.h>

// ---------------- problem constants ----------------
#define C_CLS   91
#define TOPK_N  1000
#define DETS    300
#define NBATCH  2
#define NLEV    5
#define NREG    (NBATCH * NLEV)           // 10 (batch, level) regions
#define NCAND   (NLEV * TOPK_N)           // 5000 candidates per batch
#define SB      18345600LL                // scores per batch  (HWA * C)
#define AB      201600                    // anchors per batch (HWA)
#define NEGV    (-1000000000.0f)
#define LOGIT_T (-2.9444389791664403f)    // log(0.05/0.95): sigmoid(x)>0.05 <=> x>T
#define BBOX_CLIP 4.135166556742356f      // log(1000/16)
#define IMGW    1333.0f
#define IMGH    800.0f

// ---------------- top-k machinery constants ----------------
#define NBINS     8192                    // 13-bit radix histogram
#define BIN_SHIFT 19
#define TIE_CAP   8192
#define WAVES     8                       // 256-thread blocks = 8 wave32
#define CHUNK     128                     // floats per wave async chunk (32 lanes * 16B)
#define TILE      4096                    // floats per TDM tile (16KB)

__device__ const long long g_scnt[5] = {13759200LL, 3439800LL, 859950LL, 223587LL, 63063LL};
__device__ const long long g_soff[5] = {0LL, 13759200LL, 17199000LL, 18058950LL, 18282537LL};
__device__ const int       g_aoff[5] = {0, 151200, 189000, 198450, 200907};

// ---------------- workspace layout (bytes) ----------------
#define WS_HIST      0            // 10 * 8192 * 4       = 327680
#define WS_CUTBIN    327680       // 10 * 4
#define WS_CUTNEED   327720       // 10 * 4
#define WS_ABOVECNT  327760       // 10 * 4
#define WS_TIECNT    327800       // 10 * 4
#define WS_ABOVE     327840       // 10 * 1024 * 8       = 81920
#define WS_TIES      409760       // 10 * 8192 * 8       = 655360
#define WS_TOPK      1065120      // 10 * 1024 * 8       = 81920
#define WS_CBOX      1147040      // 2 * 5000 * 4 * 4    = 160000
#define WS_CSCORE    1307040      // 2 * 5000 * 4        = 40000
#define WS_CLABEL    1347040      // 2 * 5000 * 4        = 40000
#define WS_MAXC      1387040      // 4
#define WS_KEEP      1387044      // 2 * 300 * 4         = 2400

// ---------------- helpers ----------------
__device__ __forceinline__ uint32_t fkey(float f) {
  uint32_t b = __float_as_uint(f);
  return (b & 0x80000000u) ? ~b : (b | 0x80000000u);   // monotone float->uint
}

// ----- CDNA5 async global->LDS copy (16B per lane, 512B per wave32) -----
__device__ __forceinline__ void async_ld_b128(uint32_t lds_off, const void* g) {
  asm volatile("global_load_async_to_lds_b128 %0, %1, off"
               :: "v"(lds_off), "v"(g) : "memory");
}
__device__ __forceinline__ void wait_async0() { asm volatile("s_wait_asynccnt 0x0" ::: "memory"); }
__device__ __forceinline__ void wait_async1() { asm volatile("s_wait_asynccnt 0x1" ::: "memory"); }
__device__ __forceinline__ void wait_ds0()    { asm volatile("s_wait_dscnt 0x0"    ::: "memory"); }

// ----- CDNA5 Tensor Data Mover: 1D tile (n fp32) global -> LDS -----
typedef unsigned int v4u_t __attribute__((ext_vector_type(4)));
typedef int          v8i_t __attribute__((ext_vector_type(8)));
typedef int          v4i_t __attribute__((ext_vector_type(4)));

// D# per CDNA5 ISA. group0: count=1(user), lds_addr[63:32],
// global_addr[120:64], type=2 ("image") [127:126]. group1: data_size=4B ([17:16]=2),
// tensor_dim0 [79:48] = tile_dim0 [127:112] = n, tensor_dim1 [111:80] = tile_dim1
// [143:128] = 1, tensor_dim0_stride [207:160] = n. Groups 2/3 zero (<=2D tensor).
__device__ __forceinline__ void tdm_load_1d(uint32_t lds_off, const void* g, int n) {
  const unsigned long long ga = (unsigned long long)(size_t)g;
  v4u_t g0;
  g0.x = 0x1u;                                            // count=1, user descriptor
  g0.y = lds_off;                                         // lds_addr
  g0.z = (unsigned)ga;                                    // global_addr[31:0]
  g0.w = ((unsigned)(ga >> 32) & 0x01FFFFFFu) | 0x80000000u; // ga[56:32] | type=2
  v8i_t g1;
  g1[0] = (int)(0x2u << 16);                              // wg_mask=0, data_size=4B
  g1[1] = (int)(((unsigned)n & 0xFFFFu) << 16);           // tensor_dim0[15:0] @ bits 63:48
  g1[2] = (int)(((unsigned)n >> 16) | (1u << 16));        // tensor_dim0[31:16] | tensor_dim1=1
  g1[3] = (int)(((unsigned)n & 0xFFFFu) << 16);           // tile_dim0 @ bits 127:112
  g1[4] = 1;                                              // tile_dim1=1, tile_dim2=0
  g1[5] = n;                                              // tensor_dim0_stride[31:0]
  g1[6] = 0;                                              // stride hi / dim1_stride lo
  g1[7] = 0;
  const v4i_t gz = {0, 0, 0, 0};
#if defined(__clang_major__) && (__clang_major__ >= 23)
  const v8i_t gz8 = {0, 0, 0, 0, 0, 0, 0, 0};
  __builtin_amdgcn_tensor_load_to_lds(g0, g1, gz, gz, gz8, 0);
#else
  __builtin_amdgcn_tensor_load_to_lds(g0, g1, gz, gz, 0);
#endif
}

// Stream `cnt` floats of `base` through LDS using double-buffered per-lane async
// copies. f(value, flat_index) is called once per element. Tail (< CHUNK) direct.
template <typename F>
__device__ void stream_async(const float* __restrict__ base, long long cnt,
                             float (*stage)[2][CHUNK], F f) {
  const int wave = threadIdx.x >> 5;
  const int lane = threadIdx.x & 31;
  const long long nchunk  = cnt / CHUNK;
  const long long wstride = (long long)gridDim.x * WAVES;
  long long c = (long long)blockIdx.x * WAVES + wave;
  uint32_t ldsoff[2];
  ldsoff[0] = (uint32_t)(size_t)&stage[wave][0][lane * 4];
  ldsoff[1] = (uint32_t)(size_t)&stage[wave][1][lane * 4];
  if (c < nchunk)           async_ld_b128(ldsoff[0], base + c * CHUNK + lane * 4);
  if (c + wstride < nchunk) async_ld_b128(ldsoff[1], base + (c + wstride) * CHUNK + lane * 4);
  int buf = 0;
  for (; c < nchunk; c += wstride, buf ^= 1) {
    if (c + wstride < nchunk) wait_async1();   // 2 outstanding -> oldest (this buf) done
    else                      wait_async0();   // last in flight -> drain
    const float4 v = *(const float4*)&stage[wave][buf][lane * 4];
    const long long ib = c * CHUNK + lane * 4;
    f(v.x, ib + 0); f(v.y, ib + 1); f(v.z, ib + 2); f(v.w, ib + 3);
    const long long c2 = c + 2 * wstride;
    if (c2 < nchunk) {
      wait_ds0();  // LDS reads of this buffer retired before async engine overwrites it
      async_ld_b128(ldsoff[buf], base + c2 * CHUNK + lane * 4);
    }
  }
  const long long t0 = nchunk * CHUNK;
  for (long long i = t0 + (long long)blockIdx.x * blockDim.x + threadIdx.x;
       i < cnt; i += (long long)gridDim.x * blockDim.x)
    f(base[i], i);
}

// Descending bitonic sort of n (power of two) uint64 keys in LDS.
__device__ void bitonic_desc(unsigned long long* s, int n) {
  for (unsigned k = 2; k <= (unsigned)n; k <<= 1) {
    for (unsigned j = k >> 1; j > 0; j >>= 1) {
      __syncthreads();
      for (unsigned i = threadIdx.x; i < (unsigned)n; i += blockDim.x) {
        const unsigned ix = i ^ j;
        if (ix > i) {
          const unsigned long long a = s[i], b = s[ix];
          const bool up = ((i & k) == 0);
          if (up ? (a < b) : (a > b)) { s[i] = b; s[ix] = a; }
        }
      }
    }
  }
  __syncthreads();
}

// ---------------- kernels ----------------
__global__ void zero_kernel(uint32_t* p, long long n) {
  for (long long i = (long long)blockIdx.x * blockDim.x + threadIdx.x;
       i < n; i += (long long)gridDim.x * blockDim.x) p[i] = 0u;
}

// Pass 1: per-(batch,level) 8192-bin histogram of sortable logit keys.
// TDM pipeline: one 16KB tile per descriptor, double buffered, TENSORcnt-sync'd.
__global__ void __launch_bounds__(256) hist_kernel(const float* __restrict__ cls,
                                                   uint32_t* __restrict__ hist) {
  __shared__ uint32_t lh[NBINS];                         // 32 KB
  __shared__ __align__(16) float stage[2][TILE];         // 2 x 16 KB TDM staging
  const int r = blockIdx.y, b = r / NLEV, l = r % NLEV;
  for (int i = threadIdx.x; i < NBINS; i += blockDim.x) lh[i] = 0u;
  const long long cnt = g_scnt[l];
  const float* base = cls + (long long)b * SB + g_soff[l];
  const long long nbc = cnt / TILE;
  const long long bstride = gridDim.x;
  const uint32_t soff0 = (uint32_t)(size_t)&stage[0][0];
  const uint32_t soff1 = (uint32_t)(size_t)&stage[1][0];
  long long c = blockIdx.x;
  if (threadIdx.x == 0 && c < nbc) tdm_load_1d(soff0, base + c * TILE, TILE);
  __syncthreads();                                       // also covers lh init
  int buf = 0;
  for (; c < nbc; c += bstride, buf ^= 1) {
    if (threadIdx.x == 0) {
      const long long cn = c + bstride;
      if (cn < nbc) {
        tdm_load_1d(buf ? soff0 : soff1, base + cn * TILE, TILE);
        __builtin_amdgcn_s_wait_tensorcnt(1);            // oldest (this tile) done
      } else {
        __builtin_amdgcn_s_wait_tensorcnt(0);            // drain
      }
    }
    __syncthreads();                                     // tile visible to all waves
    const float4* sv = (const float4*)stage[buf];
    for (int i = threadIdx.x; i < TILE / 4; i += blockDim.x) {
      const float4 v = sv[i];
      uint32_t u;
      u = (v.x > LOGIT_T) ? fkey(v.x) : 0u; atomicAdd(&lh[u >> BIN_SHIFT], 1u);
      u = (v.y > LOGIT_T) ? fkey(v.y) : 0u; atomicAdd(&lh[u >> BIN_SHIFT], 1u);
      u = (v.z > LOGIT_T) ? fkey(v.z) : 0u; atomicAdd(&lh[u >> BIN_SHIFT], 1u);
      u = (v.w > LOGIT_T) ? fkey(v.w) : 0u; atomicAdd(&lh[u >> BIN_SHIFT], 1u);
    }
    __syncthreads();                                     // reads done before refill
  }
  // tail (< TILE floats) direct from global
  for (long long i = nbc * TILE + (long long)blockIdx.x * blockDim.x + threadIdx.x;
       i < cnt; i += (long long)gridDim.x * blockDim.x) {
    const float v = base[i];
    const uint32_t u = (v > LOGIT_T) ? fkey(v) : 0u;
    atomicAdd(&lh[u >> BIN_SHIFT], 1u);
  }
  __syncthreads();
  uint32_t* gh = hist + (size_t)r * NBINS;
  for (int i = threadIdx.x; i < NBINS; i += blockDim.x)
    if (lh[i]) atomicAdd(&gh[i], lh[i]);
}

// Find cutoff bin: cumulative-from-top reaches TOPK.
__global__ void scan_kernel(const uint32_t* __restrict__ hist,
                            uint32_t* cut_bin, uint32_t* cut_need) {
  const int r = blockIdx.x;
  if (threadIdx.x != 0) return;
  const uint32_t* h = hist + (size_t)r * NBINS;
  long long cum = 0; int bin = 0;
  for (int i = NBINS - 1; i >= 0; --i) {
    const long long c2 = cum + h[i];
    if (c2 >= TOPK_N) { bin = i; break; }
    cum = c2;
  }
  cut_bin[r]  = (uint32_t)bin;
  cut_need[r] = (uint32_t)(TOPK_N - cum);
}

// Pass 2: emit strictly-above-cutoff items and tie-bin items as (key32, ~idx).
__global__ void __launch_bounds__(256) collect_kernel(
    const float* __restrict__ cls, const uint32_t* __restrict__ cut_bin,
    uint32_t* above_cnt, uint32_t* tie_cnt,
    unsigned long long* __restrict__ above, unsigned long long* __restrict__ ties) {
  __shared__ __align__(16) float stage[WAVES][2][CHUNK];
  const int r = blockIdx.y, b = r / NLEV, l = r % NLEV;
  const uint32_t T = cut_bin[r];
  const float* base = cls + (long long)b * SB + g_soff[l];
  unsigned long long* av = above + (size_t)r * 1024;
  unsigned long long* tv = ties  + (size_t)r * TIE_CAP;
  stream_async(base, g_scnt[l], stage, [&](float v, long long idx) {
    const uint32_t u   = (v > LOGIT_T) ? fkey(v) : 0u;
    const uint32_t bin = u >> BIN_SHIFT;
    if (bin < T) return;
    const unsigned long long key =
        ((unsigned long long)u << 32) | (unsigned long long)(0xFFFFFFFFu - (uint32_t)idx);
    if (bin > T) {
      const uint32_t p = atomicAdd(&above_cnt[r], 1u);
      if (p < 1024u) av[p] = key;                        // guaranteed < TOPK by construction
    } else {
      const uint32_t p = atomicAdd(&tie_cnt[r], 1u);
      if (p < (uint32_t)TIE_CAP) tv[p] = key;
    }
  });
}

// Resolve tie bin (sort up to 8192 ties), assemble + fully sort the top-1000.
__global__ void __launch_bounds__(1024) resolve_kernel(
    const uint32_t* cut_need, const uint32_t* above_cnt, const uint32_t* tie_cnt,
    const unsigned long long* __restrict__ above,
    const unsigned long long* __restrict__ ties,
    unsigned long long* __restrict__ topk) {
  __shared__ unsigned long long sk[TIE_CAP];             // 64 KB (LDS = 320 KB/WGP)
  __shared__ unsigned long long fk[1024];                // 8 KB final sort buffer
  const int r = blockIdx.x;
  const int M = ((int)tie_cnt[r] < TIE_CAP) ? (int)tie_cnt[r] : TIE_CAP;
  int na = (int)above_cnt[r]; if (na > 1000) na = 1000;
  int need = (int)cut_need[r]; if (need > 1024 - na) need = 1024 - na;
  const unsigned long long* tv = ties + (size_t)r * TIE_CAP;
  for (int i = threadIdx.x; i < TIE_CAP; i += blockDim.x) sk[i] = (i < M) ? tv[i] : 0ull;
  bitonic_desc(sk, TIE_CAP);
  unsigned long long* out = topk + (size_t)r * 1024;
  const unsigned long long* av = above + (size_t)r * 1024;
  for (int i = threadIdx.x; i < na; i += blockDim.x)          out[i] = av[i];
  for (int i = threadIdx.x; i < need; i += blockDim.x)        out[na + i] = sk[i];
  for (int i = na + need + threadIdx.x; i < 1024; i += blockDim.x) out[i] = 0ull;
  __syncthreads();
  for (int i = threadIdx.x; i < 1024; i += blockDim.x) fk[i] = out[i];
  bitonic_desc(fk, 1024);
  for (int i = threadIdx.x; i < 1024; i += blockDim.x) out[i] = fk[i];
}

// Decode + clip boxes for the 10 * 1000 selections; build candidate arrays.
__global__ void decode_kernel(const float* __restrict__ cls,
                              const float* __restrict__ reg,
                              const float* __restrict__ anc,
                              const unsigned long long* __restrict__ topk,
                              float* __restrict__ cboxes, float* __restrict__ cscores,
                              int* __restrict__ clabels) {
  const int t = blockIdx.x * blockDim.x + threadIdx.x;
  if (t >= NREG * TOPK_N) return;
  const int r = t / TOPK_N, j = t % TOPK_N;
  const int b = r / NLEV, l = r % NLEV;
  const unsigned long long key = topk[(size_t)r * 1024 + j];
  const int ci = b * NCAND + l * TOPK_N + j;
  float bx0 = 0.f, bx1 = 0.f, bx2 = 0.f, bx3 = 0.f, sc = NEGV; int lab = 0;
  if (key != 0ull) {
    const uint32_t idx = 0xFFFFFFFFu - (uint32_t)(key & 0xFFFFFFFFull);
    const int al = (int)(idx / C_CLS);
    lab = (int)(idx % C_CLS);
    const float logit = cls[(long long)b * SB + g_soff[l] + idx];
    const float s = 1.0f / (1.0f + expf(-logit));
    sc = (s > 0.05f) ? s : NEGV;                         // reference re-masks post-topk
    const int ag = g_aoff[l] + al;
    const float* rg = reg + ((long long)b * AB + ag) * 4;
    const float* an = anc + (size_t)ag * 4;
    const float w  = an[2] - an[0], h  = an[3] - an[1];
    const float cx = an[0] + 0.5f * w, cy = an[1] + 0.5f * h;
    const float dw = fminf(rg[2], BBOX_CLIP), dh = fminf(rg[3], BBOX_CLIP);
    const float pcx = rg[0] * w + cx, pcy = rg[1] * h + cy;
    const float pw = expf(dw) * w, ph = expf(dh) * h;
    bx0 = fminf(fmaxf(pcx - 0.5f * pw, 0.f), IMGW);
    bx1 = fminf(fmaxf(pcy - 0.5f * ph, 0.f), IMGH);
    bx2 = fminf(fmaxf(pcx + 0.5f * pw, 0.f), IMGW);
    bx3 = fminf(fmaxf(pcy + 0.5f * ph, 0.f), IMGH);
  }
  float* o = cboxes + (size_t)ci * 4;
  o[0] = bx0; o[1] = bx1; o[2] = bx2; o[3] = bx3;
  cscores[ci] = sc; clabels[ci] = lab;
}

// Global max over all candidate boxes (max_c = boxes.max() + 1).
__global__ void __launch_bounds__(1024) maxred_kernel(const float* __restrict__ cboxes,
                                                      float* maxc) {
  __shared__ float sm[1024];
  float m = -3.402823466e38f;
  for (int i = threadIdx.x; i < NBATCH * NCAND * 4; i += blockDim.x)
    m = fmaxf(m, cboxes[i]);
  sm[threadIdx.x] = m; __syncthreads();
  for (int s = 512; s > 0; s >>= 1) {
    if (threadIdx.x < s) sm[threadIdx.x] = fmaxf(sm[threadIdx.x], sm[threadIdx.x + s]);
    __syncthreads();
  }
  if (threadIdx.x == 0) *maxc = sm[0] + 1.0f;
}

// Greedy class-offset NMS: one workgroup per batch, 5000-box working set in LDS.
__global__ void __launch_bounds__(1024) nms_kernel(
    const float* __restrict__ cboxes, const float* __restrict__ cscores,
    const int* __restrict__ clabels, const float* __restrict__ maxc,
    int* __restrict__ keep) {
  __shared__ float sx1[NCAND], sy1[NCAND], sx2[NCAND], sy2[NCAND], sar[NCAND], ssc[NCAND];
  __shared__ float rsc[1024]; __shared__ int ridx[1024];
  __shared__ float bb[5];
  const int b = blockIdx.x;
  const float mc = *maxc;
  for (int i = threadIdx.x; i < NCAND; i += blockDim.x) {
    const float* bo = cboxes + (size_t)(b * NCAND + i) * 4;
    const float off = (float)clabels[b * NCAND + i] * mc;
    const float x1 = bo[0] + off, y1 = bo[1] + off, x2 = bo[2] + off, y2 = bo[3] + off;
    sx1[i] = x1; sy1[i] = y1; sx2[i] = x2; sy2[i] = y2;
    sar[i] = (x2 - x1) * (y2 - y1);
    ssc[i] = cscores[b * NCAND + i];
  }
  __syncthreads();
  for (int it = 0; it < DETS; ++it) {
    float best = -3.402823466e38f; int bi = NCAND;       // argmax, first-index tie-break
    for (int i = threadIdx.x; i < NCAND; i += blockDim.x) {
      const float s = ssc[i];
      if (s > best || (s == best && i < bi)) { best = s; bi = i; }
    }
    rsc[threadIdx.x] = best; ridx[threadIdx.x] = bi;
    __syncthreads();
    for (int s = 512; s > 0; s >>= 1) {
      if (threadIdx.x < s) {
        const float so = rsc[threadIdx.x + s]; const int io = ridx[threadIdx.x + s];
        const float sm = rsc[threadIdx.x];     const int im = ridx[threadIdx.x];
        if (so > sm || (so == sm && io < im)) { rsc[threadIdx.x] = so; ridx[threadIdx.x] = io; }
      }
      __syncthreads();
    }
    const int k = ridx[0];
    if (threadIdx.x == 0) {
      keep[b * DETS + it] = k;
      bb[0] = sx1[k]; bb[1] = sy1[k]; bb[2] = sx2[k]; bb[3] = sy2[k]; bb[4] = sar[k];
    }
    __syncthreads();
    const float kx1 = bb[0], ky1 = bb[1], kx2 = bb[2], ky2 = bb[3], kar = bb[4];
    for (int i = threadIdx.x; i < NCAND; i += blockDim.x) {
      const float ix1 = fmaxf(kx1, sx1[i]), iy1 = fmaxf(ky1, sy1[i]);
      const float ix2 = fminf(kx2, sx2[i]), iy2 = fminf(ky2, sy2[i]);
      const float inter = fmaxf(ix2 - ix1, 0.f) * fmaxf(iy2 - iy1, 0.f);
      const float iou = inter / (kar + sar[i] - inter);
      if (iou > 0.5f) ssc[i] = NEGV;                     // suppresses chosen box too (iou=1)
    }
    __syncthreads();
  }
}

// Gather kept detections into d_out: boxes[2400] | scores[600] | labels[600].
__global__ void gather_kernel(const float* __restrict__ cboxes,
                              const float* __restrict__ cscores,
                              const int* __restrict__ clabels,
                              const int* __restrict__ keep, float* __restrict__ out) {
  const int t = blockIdx.x * blockDim.x + threadIdx.x;
  if (t >= NBATCH * DETS) return;
  const int b = t / DETS;
  const int i = keep[t];
  const float* bo = cboxes + (size_t)(b * NCAND + i) * 4;
  float* ob = out + (size_t)t * 4;
  ob[0] = bo[0]; ob[1] = bo[1]; ob[2] = bo[2]; ob[3] = bo[3];
  out[NBATCH * DETS * 4 + t] = cscores[b * NCAND + i];
  out[NBATCH * DETS * 5 + t] = (float)clabels[b * NCAND + i];
}

// ---------------- host entry ----------------
extern "C" void kernel_launch(void* const* d_in, const int* in_sizes, int n_in,
                              void* d_out, int out_size, void* d_ws, size_t ws_size,
                              hipStream_t stream) {
  const float* cls = (const float*)d_in[0];   // (2, 201600, 91)
  const float* reg = (const float*)d_in[1];   // (2, 201600, 4)
  const float* anc = (const float*)d_in[2];   // (201600, 4)
  float* out = (float*)d_out;
  char* ws = (char*)d_ws;
  uint32_t* hist      = (uint32_t*)(ws + WS_HIST);
  uint32_t* cut_bin   = (uint32_t*)(ws + WS_CUTBIN);
  uint32_t* cut_need  = (uint32_t*)(ws + WS_CUTNEED);
  uint32_t* above_cnt = (uint32_t*)(ws + WS_ABOVECNT);
  uint32_t* tie_cnt   = (uint32_t*)(ws + WS_TIECNT);
  unsigned long long* above = (unsigned long long*)(ws + WS_ABOVE);
  unsigned long long* ties  = (unsigned long long*)(ws + WS_TIES);
  unsigned long long* topk  = (unsigned long long*)(ws + WS_TOPK);
  float* cboxes  = (float*)(ws + WS_CBOX);
  float* cscores = (float*)(ws + WS_CSCORE);
  int*   clabels = (int*)(ws + WS_CLABEL);
  float* maxc    = (float*)(ws + WS_MAXC);
  int*   keep    = (int*)(ws + WS_KEEP);

  zero_kernel<<<128, 256, 0, stream>>>(hist, (WS_ABOVE - WS_HIST) / 4);
  dim3 gpass(256, NREG);
  hist_kernel<<<gpass, 256, 0, stream>>>(cls, hist);
  scan_kernel<<<NREG, 32, 0, stream>>>(hist, cut_bin, cut_need);
  collect_kernel<<<gpass, 256, 0, stream>>>(cls, cut_bin, above_cnt, tie_cnt, above, ties);
  resolve_kernel<<<NREG, 1024, 0, stream>>>(cut_need, above_cnt, tie_cnt, above, ties, topk);
  decode_kernel<<<(NREG * TOPK_N + 255) / 256, 256, 0, stream>>>(cls, reg, anc, topk,
                                                                 cboxes, cscores, clabels);
  maxred_kernel<<<1, 1024, 0, stream>>>(cboxes, maxc);
  nms_kernel<<<NBATCH, 1024, 0, stream>>>(cboxes, cscores, clabels, maxc, keep);
  gather_kernel<<<(NBATCH * DETS + 255) / 256, 256, 0, stream>>>(cboxes, cscores, clabels,
                                                                 keep, out);
  (void)in_sizes; (void)n_in; (void)out_size; (void)ws_size;
}